// WeightIOHMM_40054865003140
// MI455X (gfx1250) — compile-verified
//
#include <hip/hip_runtime.h>
#include <math.h>

#define NSTATE  10
#define NLABEL  50
#define BATCH   256
#define MAXLEN  1024
#define NROWS   (BATCH * MAXLEN)   // 262144 (t,b) rows, m = b*MAXLEN + t
#define KPAD    12                 // 10 states padded to 3 chunks of K=4
#define NPAD    64                 // 50 labels padded to 4 tiles of N=16

typedef float v2f __attribute__((ext_vector_type(2)));
typedef float v8f __attribute__((ext_vector_type(8)));

// ---- workspace layout (float offsets) ----
#define OFF_LT    0                              // 100: log_softmax(transition) rows
#define OFF_LTT   128                            // 100: its transpose
#define OFF_POUT  256                            // KPAD*NPAD = 768: softmax(output), zero-padded
#define OFF_EMIS  1024                           // NROWS*NSTATE
#define OFF_FWD   (OFF_EMIS + NROWS * NSTATE)
#define OFF_BWD   (OFF_FWD  + NROWS * NSTATE)

// ---------- small-parameter prep: log-softmax(trans), softmax(output) padded ----------
__global__ void hmm_prep(const float* __restrict__ trans,
                         const float* __restrict__ outw,
                         float* __restrict__ ws) {
  const int tid = threadIdx.x;
  if (tid < NSTATE) {
    float r[NSTATE];
    float m = -INFINITY;
#pragma unroll
    for (int j = 0; j < NSTATE; ++j) { r[j] = trans[tid * NSTATE + j]; m = fmaxf(m, r[j]); }
    float s = 0.f;
#pragma unroll
    for (int j = 0; j < NSTATE; ++j) s += __expf(r[j] - m);
    const float lse = m + __logf(s);
#pragma unroll
    for (int j = 0; j < NSTATE; ++j) {
      const float v = r[j] - lse;
      ws[OFF_LT  + tid * NSTATE + j] = v;
      ws[OFF_LTT + j * NSTATE + tid] = v;
    }
  }
  if (tid < KPAD) {
    if (tid < NSTATE) {
      const float* row = outw + tid * NLABEL;
      float m = -INFINITY;
      for (int n = 0; n < NLABEL; ++n) m = fmaxf(m, row[n]);
      float s = 0.f;
      for (int n = 0; n < NLABEL; ++n) s += __expf(row[n] - m);
      const float lse = m + __logf(s);
      for (int n = 0; n < NPAD; ++n)
        ws[OFF_POUT + tid * NPAD + n] = (n < NLABEL) ? __expf(row[n] - lse) : 0.f;
    } else {  // K-pad rows 10,11 contribute zero to the matmul
      for (int n = 0; n < NPAD; ++n) ws[OFF_POUT + tid * NPAD + n] = 0.f;
    }
  }
}

// ---------- emission log-probs: gather + log-softmax over 10 states ----------
__global__ void hmm_emis(const int* __restrict__ sent,
                         const float* __restrict__ iw,
                         float* __restrict__ ws) {
  const int g = blockIdx.x * blockDim.x + threadIdx.x;   // g = b*MAXLEN + t
  if (g >= NROWS) return;
  const int tok = sent[g];
  const float* row = iw + (size_t)tok * NSTATE;
  float r[NSTATE];
  float m = -INFINITY;
#pragma unroll
  for (int s = 0; s < NSTATE; ++s) { r[s] = row[s]; m = fmaxf(m, r[s]); }
  float sum = 0.f;
#pragma unroll
  for (int s = 0; s < NSTATE; ++s) sum += __expf(r[s] - m);
  const float lse = m + __logf(sum);
  float* e = ws + OFF_EMIS + (size_t)g * NSTATE;
#pragma unroll
  for (int s = 0; s < NSTATE; ++s) e[s] = r[s] - lse;
}

// ---------- sequential HMM scans: blocks [0,256) forward, [256,512) backward ----------
__global__ void __launch_bounds__(32) hmm_scan(float* __restrict__ ws) {
  const int blk = blockIdx.x;
  const bool back = blk >= BATCH;
  const int b = back ? blk - BATCH : blk;
  const int lane = threadIdx.x;            // lane i computes state i
  if (lane >= NSTATE) return;
  const float* lt   = ws + (back ? OFF_LTT : OFF_LT);
  const float* emis = ws + OFF_EMIS;
  float*       dst  = ws + (back ? OFF_BWD : OFF_FWD);

  float ltr[NSTATE];
#pragma unroll
  for (int j = 0; j < NSTATE; ++j) ltr[j] = lt[lane * NSTATE + j];

  const size_t base = (size_t)b * MAXLEN;
  const int t0  = back ? MAXLEN - 1 : 0;
  const int stp = back ? -1 : 1;

  float f = emis[(base + t0) * NSTATE + lane];
  dst[(base + t0) * NSTATE + lane] = f;

  for (int k = 1; k < MAXLEN; ++k) {
    const int t = t0 + stp * k;
    const size_t idx = (base + t) * NSTATE + lane;
    if (k + 16 < MAXLEN)  // gfx1250 global_prefetch_b8 down the time axis
      __builtin_prefetch(&emis[(base + (size_t)(t0 + stp * (k + 16))) * NSTATE + lane], 0, 0);
    float m = -INFINITY;
    float v[NSTATE];
#pragma unroll
    for (int j = 0; j < NSTATE; ++j) {
      v[j] = ltr[j] + __shfl(f, j, 32);    // broadcast previous state vector
      m = fmaxf(m, v[j]);
    }
    float s = 0.f;
#pragma unroll
    for (int j = 0; j < NSTATE; ++j) s += __expf(v[j] - m);
    f = m + __logf(s) + emis[idx];
    dst[idx] = f;
  }
}

// ---------- gamma + WMMA log-semiring matmul: score = log(gamma @ p_out) ----------
__global__ void __launch_bounds__(32) hmm_score(const float* __restrict__ ws,
                                                float* __restrict__ out) {
  __shared__ float sG[16 * KPAD];   // gamma tile, K-padded with zeros
  __shared__ float sL[16];          // per-row logsumexp
  const int lane = threadIdx.x;
  const int row  = lane & 15;
  const int half = lane >> 4;
  const size_t m0 = (size_t)blockIdx.x * 16;
  const float* emis = ws + OFF_EMIS;
  const float* fwd  = ws + OFF_FWD;
  const float* bwd  = ws + OFF_BWD;
  const float* pout = ws + OFF_POUT;

  // expected = f + b - e  (lane pair (row,row+16) splits the 10 states 5/5)
#pragma unroll
  for (int q = 0; q < 5; ++q) {
    const int s = half * 5 + q;
    const size_t idx = (m0 + row) * NSTATE + s;
    sG[row * KPAD + s] = fwd[idx] + bwd[idx] - emis[idx];
  }
  if (half == 0) { sG[row * KPAD + 10] = 0.f; sG[row * KPAD + 11] = 0.f; }
  __syncthreads();
  if (half == 0) {
    float m = -INFINITY;
#pragma unroll
    for (int s = 0; s < NSTATE; ++s) m = fmaxf(m, sG[row * KPAD + s]);
    float sum = 0.f;
#pragma unroll
    for (int s = 0; s < NSTATE; ++s) sum += __expf(sG[row * KPAD + s] - m);
    sL[row] = m + __logf(sum);
  }
  __syncthreads();
#pragma unroll
  for (int q = 0; q < 5; ++q) {       // gamma = softmax_s(expected) in place
    const int s = half * 5 + q;
    sG[row * KPAD + s] = __expf(sG[row * KPAD + s] - sL[row]);
  }
  __syncthreads();

  // A fragments (16x4 f32, 2 VGPRs): lanes 0-15 K={kb,kb+1}, lanes 16-31 K={kb+2,kb+3}
  v2f a0, a1, a2;
  a0.x = sG[row * KPAD + 2 * half + 0]; a0.y = sG[row * KPAD + 2 * half + 1];
  a1.x = sG[row * KPAD + 2 * half + 4]; a1.y = sG[row * KPAD + 2 * half + 5];
  a2.x = sG[row * KPAD + 2 * half + 8]; a2.y = sG[row * KPAD + 2 * half + 9];

#pragma unroll
  for (int nt = 0; nt < 4; ++nt) {
    const int n = nt * 16 + row;
    // B fragments (4x16 f32): row K striped across lanes, K-halves like A
    v2f b0, b1, b2;
    b0.x = pout[(2 * half + 0) * NPAD + n]; b0.y = pout[(2 * half + 1) * NPAD + n];
    b1.x = pout[(2 * half + 4) * NPAD + n]; b1.y = pout[(2 * half + 5) * NPAD + n];
    b2.x = pout[(2 * half + 8) * NPAD + n]; b2.y = pout[(2 * half + 9) * NPAD + n];

    v8f acc = {0.f, 0.f, 0.f, 0.f, 0.f, 0.f, 0.f, 0.f};
    acc = __builtin_amdgcn_wmma_f32_16x16x4_f32(false, a0, false, b0, (short)0, acc, false, false);
    acc = __builtin_amdgcn_wmma_f32_16x16x4_f32(false, a1, false, b1, (short)0, acc, false, false);
    acc = __builtin_amdgcn_wmma_f32_16x16x4_f32(false, a2, false, b2, (short)0, acc, false, false);

    if (n < NLABEL) {
#pragma unroll
      for (int v = 0; v < 8; ++v) {    // D: M = v + 8*half, N = n
        const size_t mr = m0 + (size_t)(v + 8 * half);
        out[mr * NLABEL + n] = __logf(acc[v]);
      }
    }
  }
}

extern "C" void kernel_launch(void* const* d_in, const int* in_sizes, int n_in,
                              void* d_out, int out_size, void* d_ws, size_t ws_size,
                              hipStream_t stream) {
  (void)in_sizes; (void)n_in; (void)out_size; (void)ws_size;
  const int*   sent = (const int*)d_in[0];
  const float* iw   = (const float*)d_in[1];
  const float* tr   = (const float*)d_in[2];
  const float* ow   = (const float*)d_in[3];
  float* out = (float*)d_out;
  float* ws  = (float*)d_ws;

  hmm_prep <<<1, 64, 0, stream>>>(tr, ow, ws);
  hmm_emis <<<NROWS / 256, 256, 0, stream>>>(sent, iw, ws);
  hmm_scan <<<2 * BATCH, 32, 0, stream>>>(ws);          // fwd + bwd concurrently
  hmm_score<<<NROWS / 16, 32, 0, stream>>>(ws, out);    // 16384 M-tiles, 12 WMMA each
}